// CannyEdgeExtractor_68977174774490
// MI455X (gfx1250) — compile-verified
//
#include <hip/hip_runtime.h>
#include <hip/hip_bf16.h>

#define CH   512
#define CW   512
#define TILE 32
#define IMGR 36   // TILE + 4 (input halo for Sobel-of-halo-mag)
#define MAGR 34   // TILE + 2 (magnitude halo for NMS)
#define WPR  16   // words per row (512 / 32)

// ---- CDNA5 async global->LDS copy (ASYNCcnt) -------------------------------
#if defined(__has_builtin)
#  if __has_builtin(__builtin_amdgcn_global_load_async_to_lds_b32) && \
      __has_builtin(__builtin_amdgcn_s_wait_asynccnt)
#    define HAVE_ASYNC_LDS 1
#  endif
#endif
#ifndef HAVE_ASYNC_LDS
#  define HAVE_ASYNC_LDS 0
#endif

// Builtin expects int* in the global (1) and LDS (3) address spaces.
typedef __attribute__((address_space(1))) int GI32;
typedef __attribute__((address_space(3))) int LI32;

// ---------------------------------------------------------------------------
// Pass 1: quantize -> Sobel -> |gx|+|gy| -> NMS -> weak/strong BITMAPS.
// One 32x32 tile per 256-thread block (8 wave32 waves). 36x36 input halo
// staged in LDS via async-to-LDS copies; 34x34 magnitude field in LDS.
// Output words are built with __ballot: in the NMS phase each wave32 owns one
// full 32-pixel row (lane == x), so the ballot mask IS the bitmap word.
// ---------------------------------------------------------------------------
__global__ __launch_bounds__(256) void canny_sobel_nms_kernel(
    const float* __restrict__ img,
    unsigned int* __restrict__ weakbm,
    unsigned int* __restrict__ strongbm)
{
    __shared__ float simg[IMGR * IMGR];
    __shared__ float smag[MAGR * MAGR];

    const int n   = blockIdx.z;
    const int ty0 = blockIdx.y * TILE;
    const int tx0 = blockIdx.x * TILE;
    const int tid = threadIdx.y * 16 + threadIdx.x;
    const float* base = img + (size_t)n * CH * CW;

    // ---- stage 36x36 input halo into LDS (replicate-clamped coords) ----
#if HAVE_ASYNC_LDS
    for (int i = tid; i < IMGR * IMGR; i += 256) {
        const int iy = i / IMGR, ix = i - iy * IMGR;
        const int gy = min(max(ty0 - 2 + iy, 0), CH - 1);
        const int gx = min(max(tx0 - 2 + ix, 0), CW - 1);
        __builtin_amdgcn_global_load_async_to_lds_b32(
            (GI32*)(base + (size_t)gy * CW + gx),
            (LI32*)&simg[i], 0, 0);
    }
    __builtin_amdgcn_s_wait_asynccnt(0);
    __syncthreads();
#else
    for (int i = tid; i < IMGR * IMGR; i += 256) {
        const int iy = i / IMGR, ix = i - iy * IMGR;
        const int gy = min(max(ty0 - 2 + iy, 0), CH - 1);
        const int gx = min(max(tx0 - 2 + ix, 0), CW - 1);
        simg[i] = base[(size_t)gy * CW + gx];
    }
    __syncthreads();
#endif

    // ---- quantize in place: clip(floor(x*255), 0, 255) ----
    for (int i = tid; i < IMGR * IMGR; i += 256) {
        float v = floorf(simg[i] * 255.0f);
        simg[i] = fminf(fmaxf(v, 0.0f), 255.0f);
    }
    __syncthreads();

    // ---- 34x34 Sobel L1 magnitude (zero outside the image, per NMS pad) ----
    for (int i = tid; i < MAGR * MAGR; i += 256) {
        const int iy = i / MAGR, ix = i - iy * MAGR;
        const int my = ty0 - 1 + iy, mx = tx0 - 1 + ix;
        float magv = 0.0f;
        if (my >= 0 && my < CH && mx >= 0 && mx < CW) {
            const int c = (iy + 1) * IMGR + (ix + 1);
            const float p00 = simg[c - IMGR - 1], p01 = simg[c - IMGR], p02 = simg[c - IMGR + 1];
            const float p10 = simg[c - 1],                              p12 = simg[c + 1];
            const float p20 = simg[c + IMGR - 1], p21 = simg[c + IMGR], p22 = simg[c + IMGR + 1];
            const float gxv = (p02 + 2.0f * p12 + p22) - (p00 + 2.0f * p10 + p20);
            const float gyv = (p20 + 2.0f * p21 + p22) - (p00 + 2.0f * p01 + p02);
            magv = fabsf(gxv) + fabsf(gyv);
        }
        smag[i] = magv;
    }
    __syncthreads();

    // ---- NMS + thresholds; one wave per 32-pixel row per phase ----
    const float T1 = 0.41421356237f;  // tan(22.5 deg)
    const float T2 = 2.41421356237f;  // tan(67.5 deg)
    const int lane = tid & 31;
    #pragma unroll
    for (int p = 0; p < 4; ++p) {
        const int pix = p * 256 + tid;    // 0..1023 within the tile
        const int ly  = pix >> 5;         // constant across a wave
        const int lx  = lane;             // lane == x within the row
        const int mc  = (ly + 1) * MAGR + (lx + 1);
        const float magv = smag[mc];

        const int c = (ly + 2) * IMGR + (lx + 2);
        const float p00 = simg[c - IMGR - 1], p01 = simg[c - IMGR], p02 = simg[c - IMGR + 1];
        const float p10 = simg[c - 1],                              p12 = simg[c + 1];
        const float p20 = simg[c + IMGR - 1], p21 = simg[c + IMGR], p22 = simg[c + IMGR + 1];
        const float gxv = (p02 + 2.0f * p12 + p22) - (p00 + 2.0f * p10 + p20);
        const float gyv = (p20 + 2.0f * p21 + p22) - (p00 + 2.0f * p01 + p02);
        const float ax = fabsf(gxv), ay = fabsf(gyv);

        int dy1, dx1;
        if (ay < T1 * ax)            { dy1 = 0; dx1 = 1;  }
        else if (ay >= T2 * ax)      { dy1 = 1; dx1 = 0;  }
        else if (gxv * gyv > 0.0f)   { dy1 = 1; dx1 = 1;  }
        else                         { dy1 = 1; dx1 = -1; }

        const float n1 = smag[mc + dy1 * MAGR + dx1];
        const float n2 = smag[mc - dy1 * MAGR - dx1];
        const float nms = (magv >= n1 && magv >= n2) ? magv : 0.0f;

        const unsigned long long wb = __ballot(nms > 50.0f);
        const unsigned long long sb = __ballot(nms > 150.0f);
        if (lane == 0) {
            const size_t widx = ((size_t)n * CH + (ty0 + ly)) * WPR + blockIdx.x;
            weakbm[widx]   = (unsigned int)wb;
            strongbm[widx] = (unsigned int)sb;
        }
    }
}

// ---------------------------------------------------------------------------
// Bit-parallel hysteresis: one wave32 owns a 32x32-pixel tile; lane r holds
// row r as one 32-bit word in a register. 3x3 dilation = shifts + shuffles.
// Tile iterates to its local fixed point entirely in registers; cross-tile
// coupling comes from fixed halo bits (previous state) + ping-pong launches.
// ---------------------------------------------------------------------------
__global__ __launch_bounds__(256) void canny_prop_bits_kernel(
    const unsigned int* __restrict__ ein,
    const unsigned int* __restrict__ weakbm,
    unsigned int* __restrict__ eout)
{
    const int lane = threadIdx.x;                       // 0..31 == row in tile
    const int wy   = blockIdx.y * 8 + threadIdx.y;      // tile row index 0..15
    const int wx   = blockIdx.x;                        // word column 0..15
    const int n    = blockIdx.z;
    const int y0   = wy * 32;

    const unsigned int* be = ein    + (size_t)n * CH * WPR;
    const unsigned int* bw = weakbm + (size_t)n * CH * WPR;

    const int rowi = (y0 + lane) * WPR + wx;
    unsigned int e = be[rowi];
    const unsigned int w = bw[rowi];
    const unsigned int lbit = (wx > 0)       ? (be[rowi - 1] >> 31) & 1u : 0u;
    const unsigned int rbit = (wx < WPR - 1) ? (be[rowi + 1] & 1u)       : 0u;

    // fixed horizontal-spread of the rows just above/below the tile
    unsigned int uspread = 0u, dspread = 0u;
    if (y0 > 0) {
        const int ui = (y0 - 1) * WPR + wx;
        const unsigned int u  = be[ui];
        const unsigned int ul = (wx > 0)       ? (be[ui - 1] >> 31) & 1u : 0u;
        const unsigned int ur = (wx < WPR - 1) ? (be[ui + 1] & 1u)       : 0u;
        uspread = u | (u << 1) | (u >> 1) | ul | (ur << 31);
    }
    if (y0 + 32 < CH) {
        const int di = (y0 + 32) * WPR + wx;
        const unsigned int d  = be[di];
        const unsigned int dl = (wx > 0)       ? (be[di - 1] >> 31) & 1u : 0u;
        const unsigned int dr = (wx < WPR - 1) ? (be[di + 1] & 1u)       : 0u;
        dspread = d | (d << 1) | (d >> 1) | dl | (dr << 31);
    }

    for (int it = 0; it < 64; ++it) {
        unsigned int sp = e | (e << 1) | (e >> 1) | lbit | (rbit << 31);
        unsigned int spu = __shfl_up(sp, 1);
        unsigned int spd = __shfl_down(sp, 1);
        if (lane == 0)  spu = uspread;
        if (lane == 31) spd = dspread;
        const unsigned int ne = e | (w & (sp | spu | spd));
        const unsigned long long ch = __ballot(ne != e);
        e = ne;
        if (ch == 0ull) break;
    }

    eout[rowi] = e;
}

// ---------------------------------------------------------------------------
// Finalize: bitmap -> fp32 output, 4 pixels (one nibble) per thread.
// ---------------------------------------------------------------------------
__global__ __launch_bounds__(256) void canny_final_bits_kernel(
    const unsigned int* __restrict__ bm, float* __restrict__ out, int total4)
{
    const int p = blockIdx.x * blockDim.x + threadIdx.x;
    if (p < total4) {
        const unsigned int wv = bm[p >> 3];
        const unsigned int bits = (wv >> ((p & 7) * 4)) & 0xFu;
        float4 f;
        f.x = (bits & 1u) ? 1.0f : 0.0f;
        f.y = (bits & 2u) ? 1.0f : 0.0f;
        f.z = (bits & 4u) ? 1.0f : 0.0f;
        f.w = (bits & 8u) ? 1.0f : 0.0f;
        ((float4*)out)[p] = f;
    }
}

extern "C" void kernel_launch(void* const* d_in, const int* in_sizes, int n_in,
                              void* d_out, int out_size, void* d_ws, size_t ws_size,
                              hipStream_t stream) {
    const float* img = (const float*)d_in[0];
    const int plane = CH * CW;
    const int N = in_sizes[0] / plane;               // 32*3 = 96 planes
    const size_t npix   = (size_t)N * plane;
    const size_t nwords = (size_t)N * CH * WPR;      // 1 bit / pixel

    unsigned int* weakbm = (unsigned int*)d_ws;
    unsigned int* eA     = weakbm + nwords;
    unsigned int* eB     = eA + nwords;

    dim3 block1(16, 16, 1);
    dim3 grid1(CW / TILE, CH / TILE, N);
    canny_sobel_nms_kernel<<<grid1, block1, 0, stream>>>(img, weakbm, eA);

    // Fixed (graph-capturable) propagation rounds; each round closes every
    // 32x32 tile to its local fixed point (register-resident, wave32-wide),
    // so each launch advances connectivity by up to a full tile.
    dim3 blockP(32, 8, 1);
    dim3 gridP(WPR, CH / TILE / 8, N);
    const int ITER = 32;                              // even -> result in eA
    for (int k = 0; k < ITER; ++k) {
        const unsigned int* src = (k & 1) ? eB : eA;
        unsigned int*       dst = (k & 1) ? eA : eB;
        canny_prop_bits_kernel<<<gridP, blockP, 0, stream>>>(src, weakbm, dst);
    }

    const int total4 = (int)(npix / 4);
    canny_final_bits_kernel<<<(total4 + 255) / 256, 256, 0, stream>>>(
        eA, (float*)d_out, total4);
}